// GATLayer_7318624273062
// MI455X (gfx1250) — compile-verified
//
#include <hip/hip_runtime.h>
#include <math.h>

// Problem constants (match reference)
#define B_ 8
#define N_ 2048
#define F_IN_ 128
#define OUT_ 256
#define H_ 4
#define D_ 64
#define NEG_SLOPE_ 0.2f

typedef __attribute__((ext_vector_type(2))) float v2f;
typedef __attribute__((ext_vector_type(8))) float v8f;

// ---------------------------------------------------------------------------
// Kernel 1: Xs = X @ W.T  via V_WMMA_F32_16X16X4_F32 (one wave per 16x16 tile)
//   M = B*N = 16384, N = OUT_ = 256, K = F_IN_ = 128  (32 WMMA steps of K=4)
// A-matrix (16x4 f32): lane l -> row m=l&15, K = vgpr + 2*(l>>4)
// B-matrix (4x16 f32): vgpr v, lane l -> K = v + 2*(l>>4), col n = l&15
// C/D (16x16 f32):     vgpr r, lane l -> row = r + 8*(l>>4), col n = l&15
// ---------------------------------------------------------------------------
__global__ __launch_bounds__(256) void gemm_xw_wmma(const float* __restrict__ X,
                                                    const float* __restrict__ W,
                                                    float* __restrict__ Xs) {
  const int tilesN = OUT_ / 16;                       // 16
  int wave = blockIdx.x * 8 + (threadIdx.x >> 5);     // global tile id
  int tm = wave / tilesN;                             // 0..1023
  int tn = wave - tm * tilesN;                        // 0..15
  int lane = threadIdx.x & 31;
  int lh = lane >> 4;                                 // half-wave select
  int lm = lane & 15;

  const float* Xrow = X + (size_t)(tm * 16 + lm) * F_IN_;
  const float* Wrow = W + (size_t)(tn * 16 + lm) * F_IN_;

  v8f acc = {0.f, 0.f, 0.f, 0.f, 0.f, 0.f, 0.f, 0.f};
#pragma unroll
  for (int k0 = 0; k0 < F_IN_; k0 += 4) {
    int k = k0 + 2 * lh;                              // 8-byte aligned
    v2f av = *(const v2f*)(Xrow + k);                 // A: X[m, k..k+1]
    v2f bv = *(const v2f*)(Wrow + k);                 // B: W[n, k..k+1] == B[k, n]
    acc = __builtin_amdgcn_wmma_f32_16x16x4_f32(
        /*neg_a=*/false, av, /*neg_b=*/false, bv,
        /*c_mod=*/(short)0, acc, /*reuse_a=*/false, /*reuse_b=*/false);
  }

  int mbase = tm * 16 + lh * 8;
  int n = tn * 16 + lm;
#pragma unroll
  for (int r = 0; r < 8; ++r)
    Xs[(size_t)(mbase + r) * OUT_ + n] = acc[r];
}

// ---------------------------------------------------------------------------
// Kernel 2: per-(row,head) dot products with a_src / a_dst.
//   s [b][n][h] = sum_k Xs[b,n,h*64+k] * a_src[h,k]
//   dT[b][h][n] = sum_k Xs[b,n,h*64+k] * a_dst[h,k]   (transposed for LDS copy)
// ---------------------------------------------------------------------------
__global__ __launch_bounds__(256) void head_dots(const float* __restrict__ Xs,
                                                 const float* __restrict__ a_src,
                                                 const float* __restrict__ a_dst,
                                                 float* __restrict__ s,
                                                 float* __restrict__ dT) {
  int idx = blockIdx.x * blockDim.x + threadIdx.x;    // 0 .. B*N*H-1
  if (idx >= B_ * N_ * H_) return;
  int h = idx & (H_ - 1);
  int row = idx >> 2;                                 // 0 .. 16383
  const float* x = Xs + (size_t)row * OUT_ + h * D_;
  const float* as = a_src + h * D_;
  const float* ad = a_dst + h * D_;
  float ss = 0.f, dd = 0.f;
#pragma unroll 8
  for (int k = 0; k < D_; ++k) {
    float xv = x[k];
    ss = fmaf(xv, as[k], ss);
    dd = fmaf(xv, ad[k], dd);
  }
  s[(size_t)row * H_ + h] = ss;
  int b = row / N_;
  int n = row - b * N_;
  dT[((size_t)b * H_ + h) * N_ + n] = dd;
}

// ---------------------------------------------------------------------------
// Kernel 3: one block (256 threads) per (b,i) row.
//  - stage dT[b] (H*N floats = 32KB) in LDS
//  - A row kept in registers (8 floats/thread), read once (134 MB total = HBM floor)
//  - pass 1: masked row max (leaky scores); pass 2: sum of exp; then diag scale.
// ---------------------------------------------------------------------------
__device__ __forceinline__ float leaky(float v) {
  return v >= 0.f ? v : NEG_SLOPE_ * v;
}

__global__ __launch_bounds__(256) void attn_diag(const float* __restrict__ A,
                                                 const float* __restrict__ s,
                                                 const float* __restrict__ dT,
                                                 const float* __restrict__ Xs,
                                                 float* __restrict__ out) {
  __shared__ float ldsd[H_ * N_];   // 32 KB: dT[b][h][j]
  __shared__ float red[8 * H_];     // per-wave partials
  __shared__ float bc[H_];          // broadcast (max, then diag)

  const int tid = threadIdx.x;
  const int b = blockIdx.x / N_;
  const int i = blockIdx.x - b * N_;

  // cooperative copy dT[b] -> LDS (2048 float4, 8 per thread)
  const float4* src4 = (const float4*)(dT + (size_t)b * H_ * N_);
  float4* dst4 = (float4*)ldsd;
#pragma unroll
  for (int k = 0; k < (H_ * N_ / 4) / 256; ++k)
    dst4[tid + 256 * k] = src4[tid + 256 * k];
  __syncthreads();

  float sh[H_];
#pragma unroll
  for (int h = 0; h < H_; ++h) sh[h] = s[((size_t)b * N_ + i) * H_ + h];

  // A row -> registers (coalesced, read exactly once)
  const float* Arow = A + ((size_t)b * N_ + i) * N_;
  float a[8];
#pragma unroll
  for (int k = 0; k < 8; ++k) a[k] = Arow[tid + 256 * k];

  // ---- pass 1: masked max of leaky(s_i + d_j) ----
  float mx[H_];
#pragma unroll
  for (int h = 0; h < H_; ++h) mx[h] = -INFINITY;
#pragma unroll
  for (int k = 0; k < 8; ++k) {
    int j = tid + 256 * k;
    bool on = (a[k] != 0.0f);
#pragma unroll
    for (int h = 0; h < H_; ++h) {
      float v = leaky(sh[h] + ldsd[h * N_ + j]);
      if (on) mx[h] = fmaxf(mx[h], v);
    }
  }
#pragma unroll
  for (int off = 16; off > 0; off >>= 1)
#pragma unroll
    for (int h = 0; h < H_; ++h) mx[h] = fmaxf(mx[h], __shfl_down(mx[h], off));
  int wid = tid >> 5, lane = tid & 31;
  if (lane == 0)
#pragma unroll
    for (int h = 0; h < H_; ++h) red[wid * H_ + h] = mx[h];
  __syncthreads();
  if (tid == 0) {
#pragma unroll
    for (int h = 0; h < H_; ++h) {
      float m = red[h];
      for (int w = 1; w < 8; ++w) m = fmaxf(m, red[w * H_ + h]);
      bc[h] = m;
    }
  }
  __syncthreads();
#pragma unroll
  for (int h = 0; h < H_; ++h) mx[h] = bc[h];

  // ---- pass 2: sum of A_ij * exp(v - max) ----
  float sm[H_] = {0.f, 0.f, 0.f, 0.f};
#pragma unroll
  for (int k = 0; k < 8; ++k) {
    int j = tid + 256 * k;
#pragma unroll
    for (int h = 0; h < H_; ++h) {
      float v = leaky(sh[h] + ldsd[h * N_ + j]);
      sm[h] = fmaf(a[k], __expf(v - mx[h]), sm[h]);
    }
  }
#pragma unroll
  for (int off = 16; off > 0; off >>= 1)
#pragma unroll
    for (int h = 0; h < H_; ++h) sm[h] += __shfl_down(sm[h], off);
  if (lane == 0)
#pragma unroll
    for (int h = 0; h < H_; ++h) red[wid * H_ + h] = sm[h];
  __syncthreads();
  if (tid == 0) {
#pragma unroll
    for (int h = 0; h < H_; ++h) {
      float t = red[h];
      for (int w = 1; w < 8; ++w) t += red[w * H_ + h];
      // numerator: diagonal score (self-loop guarantees mask==1, max finite)
      float vii = leaky(sh[h] + ldsd[h * N_ + i]);
      bc[h] = __expf(vii - mx[h]) / t;
    }
  }
  __syncthreads();

  // out[b,i,:] = diag[h] * Xs[b,i,:]
  float dg = bc[tid >> 6];  // 64 feature cols per head
  size_t base = ((size_t)b * N_ + i) * OUT_;
  out[base + tid] = dg * Xs[base + tid];
}

// ---------------------------------------------------------------------------
extern "C" void kernel_launch(void* const* d_in, const int* in_sizes, int n_in,
                              void* d_out, int out_size, void* d_ws, size_t ws_size,
                              hipStream_t stream) {
  const float* A     = (const float*)d_in[0];  // (B,N,N)
  const float* X     = (const float*)d_in[1];  // (B,N,F_IN)
  const float* W     = (const float*)d_in[2];  // (OUT,F_IN)
  const float* a_src = (const float*)d_in[3];  // (H,D)
  const float* a_dst = (const float*)d_in[4];  // (H,D)
  float* out = (float*)d_out;                  // (B,N,OUT)

  // workspace: Xs (16.78 MB) | s (256 KB) | dT (256 KB)
  float* Xs = (float*)d_ws;
  float* s  = Xs + (size_t)B_ * N_ * OUT_;
  float* dT = s + (size_t)B_ * N_ * H_;

  // 1) GEMM: 16384/16 * 256/16 = 16384 tiles; 8 waves/block -> 2048 blocks
  gemm_xw_wmma<<<2048, 256, 0, stream>>>(X, W, Xs);

  // 2) head dots: B*N*H = 65536 threads
  head_dots<<<(B_ * N_ * H_ + 255) / 256, 256, 0, stream>>>(Xs, a_src, a_dst, s, dT);

  // 3) softmax-diagonal + scale: one block per (b,i) row
  attn_diag<<<B_ * N_, 256, 0, stream>>>(A, s, dT, Xs, out);
}